// InstructionTracePositionEmbedding_53961969107317
// MI455X (gfx1250) — compile-verified
//
#include <hip/hip_runtime.h>
#include <stdint.h>

#define BATCH   32
#define SEQ     4096
#define HID     256
#define NEXT_ID 5
#define EPSF    1e-5f

typedef float f4  __attribute__((ext_vector_type(4)));
typedef float v2f __attribute__((ext_vector_type(2)));
typedef float v8f __attribute__((ext_vector_type(8)));
typedef int   i4v __attribute__((vector_size(16)));   // matches builtin param type

#define AS1 __attribute__((address_space(1)))
#define AS3 __attribute__((address_space(3)))

#if __has_builtin(__builtin_amdgcn_wmma_f32_16x16x4_f32)
#define HAVE_WMMA_F32X4 1
#endif
#if __has_builtin(__builtin_amdgcn_global_load_async_to_lds_b128) && \
    __has_builtin(__builtin_amdgcn_s_wait_asynccnt)
#define HAVE_ASYNC_LDS 1
#endif

// ---------------------------------------------------------------------------
// Kernel 1: per-batch-row scan. Computes instruction index (cumsum of starts)
// and argument index (pos - cummax(start?pos:0)), packs (instr<<16)|args.
// One 256-thread block per batch row, 16 elements per thread.
// ---------------------------------------------------------------------------
__global__ void __launch_bounds__(256) itpe_scan(const int* __restrict__ state,
                                                 int* __restrict__ ws)
{
    const int b   = blockIdx.x;
    const int tid = threadIdx.x;
    const int j0  = tid * 16;
    const int* row = state + b * SEQ;

    // local pass: count of starts + position of last start in my 16 elems
    unsigned mask = 0u;
    int cnt = 0, mx = 0;
    #pragma unroll
    for (int i = 0; i < 16; ++i) {
        const int j = j0 + i;
        const bool s = (j > 0) && (row[j - 1] == NEXT_ID);
        mask |= (s ? 1u : 0u) << i;
        cnt += s ? 1 : 0;
        if (s) mx = j;          // j increasing -> last start == max start pos
    }

    const int lane = tid & 31;
    const int wid  = tid >> 5;

    // wave32 inclusive scan of (cnt, mx)
    int icnt = cnt, imx = mx;
    #pragma unroll
    for (int off = 1; off < 32; off <<= 1) {
        const int c = __shfl_up(icnt, off, 32);
        const int m = __shfl_up(imx,  off, 32);
        if (lane >= off) { icnt += c; imx = (imx > m) ? imx : m; }
    }

    __shared__ int wc[8], wm[8];
    if (lane == 31) { wc[wid] = icnt; wm[wid] = imx; }
    __syncthreads();

    // exclusive cross-wave prefix (8 waves -> trivial serial)
    int ec = 0, em = 0;
    for (int w = 0; w < wid; ++w) {
        ec += wc[w];
        em = (em > wm[w]) ? em : wm[w];
    }
    // exclusive-of-self within wave
    int pc = __shfl_up(icnt, 1, 32);
    int pm = __shfl_up(imx,  1, 32);
    if (lane == 0) { pc = 0; pm = 0; }

    int run  = ec + pc;
    int last = (em > pm) ? em : pm;

    #pragma unroll
    for (int i = 0; i < 16; ++i) {
        const int j = j0 + i;
        const bool s = (mask >> i) & 1u;
        run += s ? 1 : 0;
        if (s) last = j;
        ws[b * SEQ + j] = (run << 16) | (j - last);
    }
}

// ---------------------------------------------------------------------------
// Kernel 2: fused 3-way gather + LayerNorm. One wave per output row.
// Block = 256 threads = 8 waves. Lane L owns hidden elems [8L, 8L+8).
// LN reductions done exactly in f32 on the matrix pipe via
// V_WMMA_F32_16X16X4_F32 (ones/selector B), shuffle fallback otherwise.
// gamma/beta staged to LDS via async load-to-LDS when available.
// ---------------------------------------------------------------------------
__global__ void __launch_bounds__(256) itpe_fuse(
    const int*   __restrict__ state,
    const float* __restrict__ temb,
    const float* __restrict__ iemb,
    const float* __restrict__ aemb,
    const float* __restrict__ gamma,
    const float* __restrict__ beta,
    const int*   __restrict__ ws,
    float*       __restrict__ out)
{
    __shared__ float sgb[2 * HID];   // [0,256)=gamma, [256,512)=beta
    const int tid = threadIdx.x;

#ifdef HAVE_ASYNC_LDS
    if (tid < 128) {
        const float* src = (tid < 64) ? (gamma + (size_t)tid * 4)
                                      : (beta  + (size_t)(tid - 64) * 4);
        __builtin_amdgcn_global_load_async_to_lds_b128(
            (AS1 i4v*)(AS1 void*)(void*)src,
            (AS3 i4v*)(AS3 void*)(void*)&sgb[tid * 4],
            0, 0);
    }
    __builtin_amdgcn_s_wait_asynccnt(0);
#else
    if (tid < 64)        ((f4*)sgb)[tid] = ((const f4*)gamma)[tid];
    else if (tid < 128)  ((f4*)sgb)[tid] = ((const f4*)beta)[tid - 64];
#endif
    __syncthreads();

    const int lane = tid & 31;
    const int wid  = tid >> 5;
    const int row  = blockIdx.x * 8 + wid;      // < BATCH*SEQ exactly

    const int tok    = state[row];
    const int packed = ws[row];
    const int ins    = packed >> 16;
    const int arg    = packed & 0xFFFF;
    const int h      = lane * 8;

    const f4* tp = (const f4*)(temb + (size_t)tok * HID + h);
    const f4* ip = (const f4*)(iemb + (size_t)ins * HID + h);
    const f4* ap = (const f4*)(aemb + (size_t)arg * HID + h);
    const f4 e0 = tp[0] + ip[0] + ap[0];
    const f4 e1 = tp[1] + ip[1] + ap[1];

    float s = (e0.x + e0.y) + (e0.z + e0.w) + (e1.x + e1.y) + (e1.z + e1.w);
    float sq = 0.f;
    sq = fmaf(e0.x, e0.x, sq); sq = fmaf(e0.y, e0.y, sq);
    sq = fmaf(e0.z, e0.z, sq); sq = fmaf(e0.w, e0.w, sq);
    sq = fmaf(e1.x, e1.x, sq); sq = fmaf(e1.y, e1.y, sq);
    sq = fmaf(e1.z, e1.z, sq); sq = fmaf(e1.w, e1.w, sq);

    float totS, totQ;
#ifdef HAVE_WMMA_F32X4
    // A fragment (16x4 f32, 2 VGPRs): A[m,0]=s_m, A[m,1]=sq_m,
    // A[m,2]=s_{m+16}, A[m,3]=sq_{m+16}   (m = lane%16)
    v2f a;    a[0] = s;  a[1] = sq;
    // B selector: col0 picks k={0,2} (s terms), col1 picks k={1,3} (sq terms)
    v2f bsel; bsel[0] = (lane == 0 || lane == 16) ? 1.f : 0.f;
              bsel[1] = (lane == 1 || lane == 17) ? 1.f : 0.f;
    const v8f zero = {0.f,0.f,0.f,0.f,0.f,0.f,0.f,0.f};
    // Stage 1: fold K -> D[m,0] = s_m+s_{m+16}, D[m,1] = sq_m+sq_{m+16}
    v8f d1 = __builtin_amdgcn_wmma_f32_16x16x4_f32(false, a, false, bsel,
                                                   (short)0, zero, false, false);
    // Stage 2: fold M with A = ones(16x4); B_i rows come from D rows
    // {2i, 2i+1, 2i+8, 2i+9} -> accumulated column sums of D.
    v2f onesA; onesA[0] = 1.f; onesA[1] = 1.f;
    v8f d2 = zero;
    v2f bb;
    bb[0] = d1[0]; bb[1] = d1[1];
    d2 = __builtin_amdgcn_wmma_f32_16x16x4_f32(false, onesA, false, bb,
                                               (short)0, d2, false, false);
    bb[0] = d1[2]; bb[1] = d1[3];
    d2 = __builtin_amdgcn_wmma_f32_16x16x4_f32(false, onesA, false, bb,
                                               (short)0, d2, false, false);
    bb[0] = d1[4]; bb[1] = d1[5];
    d2 = __builtin_amdgcn_wmma_f32_16x16x4_f32(false, onesA, false, bb,
                                               (short)0, d2, false, false);
    bb[0] = d1[6]; bb[1] = d1[7];
    d2 = __builtin_amdgcn_wmma_f32_16x16x4_f32(false, onesA, false, bb,
                                               (short)0, d2, false, false);
    // lane 0 column = total s, lane 1 column = total sq (exact f32 sums)
    totS = __shfl(d2[0], 0, 32);
    totQ = __shfl(d2[0], 1, 32);
#else
    float ls = s, lq = sq;
    #pragma unroll
    for (int off = 16; off; off >>= 1) {
        ls += __shfl_xor(ls, off, 32);
        lq += __shfl_xor(lq, off, 32);
    }
    totS = ls; totQ = lq;
#endif

    const float mean = totS * (1.0f / HID);
    const float var  = fmaf(totQ, 1.0f / HID, -(mean * mean));
    const float rstd = rsqrtf(var + EPSF);

    const f4 g0 = *(const f4*)(sgb + h);
    const f4 g1 = *(const f4*)(sgb + h + 4);
    const f4 b0 = *(const f4*)(sgb + HID + h);
    const f4 b1 = *(const f4*)(sgb + HID + h + 4);

    f4 o0, o1;
    o0.x = fmaf((e0.x - mean) * rstd, g0.x, b0.x);
    o0.y = fmaf((e0.y - mean) * rstd, g0.y, b0.y);
    o0.z = fmaf((e0.z - mean) * rstd, g0.z, b0.z);
    o0.w = fmaf((e0.w - mean) * rstd, g0.w, b0.w);
    o1.x = fmaf((e1.x - mean) * rstd, g1.x, b1.x);
    o1.y = fmaf((e1.y - mean) * rstd, g1.y, b1.y);
    o1.z = fmaf((e1.z - mean) * rstd, g1.z, b1.z);
    o1.w = fmaf((e1.w - mean) * rstd, g1.w, b1.w);

    f4* op = (f4*)(out + (size_t)row * HID + h);
    __builtin_nontemporal_store(o0, op);       // 134 MB stream: NT policy,
    __builtin_nontemporal_store(o1, op + 1);   // keep tables resident in L2
}

// ---------------------------------------------------------------------------
extern "C" void kernel_launch(void* const* d_in, const int* in_sizes, int n_in,
                              void* d_out, int out_size, void* d_ws, size_t ws_size,
                              hipStream_t stream)
{
    const int*   state = (const int*)  d_in[0];
    const float* temb  = (const float*)d_in[1];
    const float* iemb  = (const float*)d_in[2];
    const float* aemb  = (const float*)d_in[3];
    const float* gamma = (const float*)d_in[4];
    const float* beta  = (const float*)d_in[5];
    int*   ws  = (int*)d_ws;                    // BATCH*SEQ packed int32 = 512 KB
    float* out = (float*)d_out;

    itpe_scan<<<dim3(BATCH), dim3(256), 0, stream>>>(state, ws);
    itpe_fuse<<<dim3((BATCH * SEQ) / 8), dim3(256), 0, stream>>>(
        state, temb, iemb, aemb, gamma, beta, ws, out);
}